// SparseAttention_49185965473951
// MI455X (gfx1250) — compile-verified
//
#include <hip/hip_runtime.h>
#include <math.h>

typedef __attribute__((ext_vector_type(2))) float v2f;
typedef __attribute__((ext_vector_type(8))) float v8f;

#define HH 4
#define EE 32
#define DD 32
#define SOFTMAX_SCALE 0.17677669529663687f  // 1/sqrt(32)

// Monotonic order-preserving float->uint encoding (atomic-max friendly).
// encode: non-neg -> bits|0x80000000 ; neg -> ~bits.  0u is below every real encoding.
__device__ __forceinline__ unsigned enc_f32(float f) {
  unsigned u = __float_as_uint(f);
  return (u & 0x80000000u) ? ~u : (u | 0x80000000u);
}
__device__ __forceinline__ float dec_f32(unsigned e) {
  unsigned u = (e & 0x80000000u) ? (e ^ 0x80000000u) : ~e;
  return __uint_as_float(u);
}

// ---------------------------------------------------------------------------
// Pass 1: per-16-edge tile, per head: scores = diag( Qg (16x32) * Kg^T (32x16) )
// via 8 chained V_WMMA_F32_16X16X4_F32; store qk, atomic-max into segMax.
// ---------------------------------------------------------------------------
__global__ void attn_scores(const float* __restrict__ Q,
                            const float* __restrict__ Kmat,
                            const int* __restrict__ eDst,
                            const int* __restrict__ eSrc,
                            float* __restrict__ qk,          // [NE*HH]
                            unsigned* __restrict__ segMax,   // [L*HH], encoded
                            int NE) {
  const int lane = threadIdx.x & 31;
  const int wavesPerBlock = blockDim.x >> 5;
  const int group = blockIdx.x * wavesPerBlock + (threadIdx.x >> 5);
  const int nGroups = (NE + 15) >> 4;
  if (group >= nGroups) return;          // wave-uniform exit: EXEC all-1s for WMMA

  const int base = group << 4;
  const int m = lane & 15;               // edge slot; lanes (m, m+16) share row m
  int e = base + m;
  if (e >= NE) e = NE - 1;               // clamp (tail); stores/atomics masked below
  const int hi = lane >> 4;              // 0: K=0,1 ; 1: K=2,3 per 4-wide chunk
  const int d = eDst[e];
  const int s = eSrc[e];

  const float* qrow = Q    + (size_t)d * (HH * EE) + 2 * hi;
  const float* krow = Kmat + (size_t)s * (HH * EE) + 2 * hi;

  const int eg = base + lane;            // edge owned by this lane post-extraction

  for (int h = 0; h < HH; ++h) {
    const float* qh = qrow + h * EE;
    const float* kh = krow + h * EE;
    v8f c = {};
#pragma unroll
    for (int k = 0; k < 8; ++k) {
      v2f a = *(const v2f*)(qh + 4 * k);   // A: lane m -> row m, K = 4k+2*hi, +1
      v2f b = *(const v2f*)(kh + 4 * k);   // B operand mirrors A layout (lane = N)
      c = __builtin_amdgcn_wmma_f32_16x16x4_f32(
          /*neg_a=*/false, a, /*neg_b=*/false, b,
          /*c_mod=*/(short)0, c, /*reuse_a=*/false, /*reuse_b=*/false);
    }
    // Diagonal extraction:
    //   m<8 : diag[m] in c[m]   at lane m
    //   m>=8: diag[m] in c[m-8] at lane m+16
    int r = (lane < 8) ? lane : (lane - 24);
    float sel = c[0];
    if (r == 1) sel = c[1];
    if (r == 2) sel = c[2];
    if (r == 3) sel = c[3];
    if (r == 4) sel = c[4];
    if (r == 5) sel = c[5];
    if (r == 6) sel = c[6];
    if (r == 7) sel = c[7];
    int srcLane = (lane < 8) ? lane : (lane + 16);
    float score = __shfl(sel, srcLane, 32) * SOFTMAX_SCALE;

    if (lane < 16 && eg < NE) {
      qk[(size_t)eg * HH + h] = score;
      __hip_atomic_fetch_max(&segMax[(size_t)d * HH + h], enc_f32(score),
                             __ATOMIC_RELAXED, __HIP_MEMORY_SCOPE_AGENT);
    }
  }
}

// ---------------------------------------------------------------------------
// Pass 2: ex = exp(qk - segMax[dst]); segSum[dst] += ex  (all 4 heads / thread)
// ---------------------------------------------------------------------------
__global__ void attn_softmax(const int* __restrict__ eDst,
                             float* __restrict__ qk,            // in scores / out ex
                             const unsigned* __restrict__ segMax,
                             float* __restrict__ segSum,
                             int NE) {
  int e = blockIdx.x * blockDim.x + threadIdx.x;
  if (e >= NE) return;
  const int d = eDst[e];

  float4 q = *(const float4*)(qk + (size_t)e * HH);
  uint4 me = *(const uint4*)(segMax + (size_t)d * HH);

  float4 x;
  x.x = __expf(q.x - dec_f32(me.x));
  x.y = __expf(q.y - dec_f32(me.y));
  x.z = __expf(q.z - dec_f32(me.z));
  x.w = __expf(q.w - dec_f32(me.w));
  *(float4*)(qk + (size_t)e * HH) = x;

  float* ss = segSum + (size_t)d * HH;
  __hip_atomic_fetch_add(&ss[0], x.x, __ATOMIC_RELAXED, __HIP_MEMORY_SCOPE_AGENT);
  __hip_atomic_fetch_add(&ss[1], x.y, __ATOMIC_RELAXED, __HIP_MEMORY_SCOPE_AGENT);
  __hip_atomic_fetch_add(&ss[2], x.z, __ATOMIC_RELAXED, __HIP_MEMORY_SCOPE_AGENT);
  __hip_atomic_fetch_add(&ss[3], x.w, __ATOMIC_RELAXED, __HIP_MEMORY_SCOPE_AGENT);
}

// ---------------------------------------------------------------------------
// Pass 3: out[dst,h,:] += (ex/segSum[dst,h]) * V[src,h,:]  — one wave per edge,
// lanes cover D=32, 4 native f32 atomic adds per lane per edge.
// ---------------------------------------------------------------------------
__global__ void attn_aggregate(const int* __restrict__ eDst,
                               const int* __restrict__ eSrc,
                               const float* __restrict__ ex,     // [NE*HH]
                               const float* __restrict__ segSum, // [L*HH]
                               const float* __restrict__ V,
                               float* __restrict__ out,
                               int NE) {
  const int lane = threadIdx.x & 31;
  const int wavesPerBlock = blockDim.x >> 5;
  const int wave = blockIdx.x * wavesPerBlock + (threadIdx.x >> 5);
  const int nWaves = gridDim.x * wavesPerBlock;

  for (int e = wave; e < NE; e += nWaves) {
    const int d = eDst[e];
    const int s = eSrc[e];
    float4 xv = *(const float4*)(ex + (size_t)e * HH);
    float4 sv = *(const float4*)(segSum + (size_t)d * HH);
    float alpha0 = xv.x / sv.x;
    float alpha1 = xv.y / sv.y;
    float alpha2 = xv.z / sv.z;
    float alpha3 = xv.w / sv.w;

    const float* vrow = V + (size_t)s * (HH * DD);
    float* orow = out + (size_t)d * (HH * DD);

    __hip_atomic_fetch_add(&orow[0 * DD + lane], alpha0 * vrow[0 * DD + lane],
                           __ATOMIC_RELAXED, __HIP_MEMORY_SCOPE_AGENT);
    __hip_atomic_fetch_add(&orow[1 * DD + lane], alpha1 * vrow[1 * DD + lane],
                           __ATOMIC_RELAXED, __HIP_MEMORY_SCOPE_AGENT);
    __hip_atomic_fetch_add(&orow[2 * DD + lane], alpha2 * vrow[2 * DD + lane],
                           __ATOMIC_RELAXED, __HIP_MEMORY_SCOPE_AGENT);
    __hip_atomic_fetch_add(&orow[3 * DD + lane], alpha3 * vrow[3 * DD + lane],
                           __ATOMIC_RELAXED, __HIP_MEMORY_SCOPE_AGENT);
  }
}

// ---------------------------------------------------------------------------
extern "C" void kernel_launch(void* const* d_in, const int* in_sizes, int n_in,
                              void* d_out, int out_size, void* d_ws, size_t ws_size,
                              hipStream_t stream) {
  const float* Q = (const float*)d_in[0];   // (L, H, E) f32
  const float* K = (const float*)d_in[1];   // (K, H, E) f32
  const float* V = (const float*)d_in[2];   // (K, H, D) f32
  const int* edges = (const int*)d_in[3];   // (2, NE) int

  const int NE = in_sizes[3] / 2;
  const int L = in_sizes[0] / (HH * EE);
  const int* eDst = edges;
  const int* eSrc = edges + NE;
  float* out = (float*)d_out;

  // Workspace layout: qk/ex [NE*4] f32 | segMax [L*4] u32 | segSum [L*4] f32
  char* ws = (char*)d_ws;
  float* qk = (float*)ws;
  unsigned* segMax = (unsigned*)(ws + (size_t)NE * HH * sizeof(float));
  float* segSum = (float*)((char*)segMax + (size_t)L * HH * sizeof(unsigned));

  hipMemsetAsync(out, 0, (size_t)out_size * sizeof(float), stream);
  hipMemsetAsync(segMax, 0, (size_t)L * HH * sizeof(unsigned), stream); // enc(-inf) floor
  hipMemsetAsync(segSum, 0, (size_t)L * HH * sizeof(float), stream);

  const int nGroups = (NE + 15) / 16;     // 16 edges per wave
  const int wavesPerBlock = 8;            // 256 threads = 8 wave32
  const int blocks1 = (nGroups + wavesPerBlock - 1) / wavesPerBlock;
  attn_scores<<<blocks1, 256, 0, stream>>>(Q, K, eDst, eSrc, qk, segMax, NE);

  const int blocks2 = (NE + 255) / 256;
  attn_softmax<<<blocks2, 256, 0, stream>>>(eDst, qk, segMax, segSum, NE);

  const int blocks3 = 1024;               // grid-stride, 8 waves/block
  attn_aggregate<<<blocks3, 256, 0, stream>>>(eDst, eSrc, qk, segSum, V, out, NE);
}